// DualPrimalEdgePooling_89318139887617
// MI455X (gfx1250) — compile-verified
//
#include <hip/hip_runtime.h>
#include <stdint.h>

typedef unsigned long long u64;
typedef unsigned int u32;
typedef float v2f __attribute__((ext_vector_type(2)));
typedef float v8f __attribute__((ext_vector_type(8)));

static constexpr int kN = 200000;      // primal nodes
static constexpr int kE = 600000;      // edges (= dual nodes)
static constexpr int kF = 128;         // feature dim
static constexpr int kKeep = 300000;
static constexpr int kPool = kE - kKeep;
static constexpr int kIters = 40;
static constexpr u32 kP2E = 1u << 20;  // pow2 >= E
static constexpr u32 kP2N = 1u << 18;  // pow2 >= N
#define PADITEM 0xFFFFFFFFFFFFFFFFull

static inline u32 cdiv(long long a, long long b) { return (u32)((a + b - 1) / b); }

// ---------------- zero-init ----------------
__global__ void zeroF(float* p, long long n) {
  long long i = (long long)blockIdx.x * blockDim.x + threadIdx.x;
  if (i < n) p[i] = 0.0f;
}
__global__ void zeroU32(u32* p, long long n) {
  long long i = (long long)blockIdx.x * blockDim.x + threadIdx.x;
  if (i < n) p[i] = 0u;
}
__global__ void zeroU8(unsigned char* p, long long n) {
  long long i = (long long)blockIdx.x * blockDim.x + threadIdx.x;
  if (i < n) p[i] = 0;
}

// ---------------- top-k selection ----------------
__global__ void attKeysK(const float* __restrict__ att, u64* __restrict__ items) {
  u32 e = blockIdx.x * blockDim.x + threadIdx.x;
  if (e >= kP2E) return;
  if (e < (u32)kE) {
    u32 b = __float_as_uint(att[e]);
    u32 m = (b & 0x80000000u) ? ~b : (b | 0x80000000u);  // monotone ascending
    u32 k = ~m;                                          // ascending key == descending att
    items[e] = ((u64)k << 32) | (u64)e;
  } else {
    items[e] = PADITEM;
  }
}

__global__ void bitonicStepK(u64* __restrict__ d, u32 j, u32 k) {
  u32 i = blockIdx.x * blockDim.x + threadIdx.x;
  u32 ixj = i ^ j;
  if (ixj > i) {
    u64 a = d[i], b = d[ixj];
    bool up = ((i & k) == 0);
    bool sw = up ? (a > b) : (a < b);
    if (sw) { d[i] = b; d[ixj] = a; }
  }
}

__global__ void poolMaskK(const u64* __restrict__ items, unsigned char* __restrict__ pm) {
  u32 p = blockIdx.x * blockDim.x + threadIdx.x;
  if (p < (u32)kPool) {
    u32 e = (u32)(items[p] & 0xFFFFFFFFull);
    pm[e] = 1;
  }
}

// ---------------- union-find / label propagation ----------------
__global__ void initLabelsK(int* __restrict__ l) {
  int i = blockIdx.x * blockDim.x + threadIdx.x;
  if (i < kN) l[i] = i;
}
__global__ void relaxK(int* __restrict__ l, const int* __restrict__ src,
                       const int* __restrict__ dst, const unsigned char* __restrict__ pm) {
  int e = blockIdx.x * blockDim.x + threadIdx.x;
  if (e >= kE) return;
  int s = src[e];
  int d = pm[e] ? dst[e] : s;   // non-pooled edges become self-edges
  int ls = l[s], ld = l[d];
  atomicMin(&l[s], ld);
  atomicMin(&l[d], ls);
}
__global__ void jumpK(int* __restrict__ out, const int* __restrict__ in) {
  int i = blockIdx.x * blockDim.x + threadIdx.x;
  if (i < kN) out[i] = in[in[i]];
}

// ---------------- unique(labels) inverse via root-rank scan ----------------
__global__ void isRootK(const int* __restrict__ l, int* __restrict__ flags) {
  int i = blockIdx.x * blockDim.x + threadIdx.x;
  if (i < kN) flags[i] = (l[i] == i) ? 1 : 0;
}

__global__ __launch_bounds__(1024) void scanBlocksK(const int* __restrict__ in,
                                                    int* __restrict__ out,
                                                    int* __restrict__ bsums, int n) {
  __shared__ int s[1024];
  int t = threadIdx.x;
  int i = blockIdx.x * 1024 + t;
  int v = (i < n) ? in[i] : 0;
  s[t] = v;
  __syncthreads();
  for (int off = 1; off < 1024; off <<= 1) {
    int add = (t >= off) ? s[t - off] : 0;
    __syncthreads();
    s[t] += add;
    __syncthreads();
  }
  if (i < n) out[i] = s[t];
  if (t == 1023) bsums[blockIdx.x] = s[1023];
}
__global__ __launch_bounds__(1024) void scanSumsExclK(int* __restrict__ sums, int nb) {
  __shared__ int s[1024];
  int t = threadIdx.x;
  int v = (t < nb) ? sums[t] : 0;
  s[t] = v;
  __syncthreads();
  for (int off = 1; off < 1024; off <<= 1) {
    int add = (t >= off) ? s[t - off] : 0;
    __syncthreads();
    s[t] += add;
    __syncthreads();
  }
  if (t < nb) sums[t] = s[t] - v;  // exclusive
}
__global__ __launch_bounds__(1024) void scanAddK(int* __restrict__ out,
                                                 const int* __restrict__ sums, int n) {
  int i = blockIdx.x * 1024 + threadIdx.x;
  if (i < n) out[i] += sums[i >> 10];
}

__global__ void clusterK(const int* __restrict__ l, const int* __restrict__ incl,
                         int* __restrict__ cluster) {
  int i = blockIdx.x * blockDim.x + threadIdx.x;
  if (i < kN) cluster[i] = incl[l[i]] - 1;  // rank of root among sorted roots
}

// ---------------- primal aggregation (WMMA one-hot segment sum) ----------------
__global__ void nodeKeysK(const int* __restrict__ cluster, u64* __restrict__ items) {
  u32 i = blockIdx.x * blockDim.x + threadIdx.x;
  if (i >= kP2N) return;
  items[i] = (i < (u32)kN) ? (((u64)(u32)cluster[i] << 32) | (u64)i) : PADITEM;
}

__global__ __launch_bounds__(256)
void primalAggWmmaK(const u64* __restrict__ items, const float* __restrict__ primalX,
                    float* __restrict__ outPX) {
  __shared__ int sClust[8][16];
  __shared__ int sNode[8][16];
  __shared__ int sSeg[8][16];
  __shared__ int sSegCl[8][16];
  const int wave = threadIdx.x >> 5;
  const int lane = threadIdx.x & 31;
  const int tile = blockIdx.x * 8 + wave;
  if (lane < 16) {
    u64 it = items[(size_t)tile * 16 + lane];
    sClust[wave][lane] = (int)(u32)(it >> 32);  // pad -> -1
    sNode[wave][lane] = (int)(u32)it;
    sSegCl[wave][lane] = -1;
  }
  __syncthreads();
  if (lane < 16) {
    int s = 0;
    for (int m = 1; m <= lane; ++m) s += (sClust[wave][m] != sClust[wave][m - 1]) ? 1 : 0;
    sSeg[wave][lane] = s;
  }
  __syncthreads();
  if (lane < 16) {
    bool head = (lane == 0) || (sClust[wave][lane] != sClust[wave][lane - 1]);
    if (head) sSegCl[wave][sSeg[wave][lane]] = sClust[wave][lane];
  }
  __syncthreads();

  // ---- hoist ALL per-tile metadata out of the feature loop (LDS -> VGPRs) ----
  // A-matrix (16x4 f32): lanes 0-15 hold K=0,1 in regs 0,1; lanes 16-31 hold K=2,3.
  const int M = lane & 15;
  const int kb = (lane >> 4) << 1;  // 0 or 2
  const int n = lane & 15;
  const int rbase = (lane >> 4) << 3;  // C/D: VGPR j holds row M = j + 8*(lane>=16)

  v2f afrag[4];
  float mskA[4], mskB[4];
  const float* rowA[4];
  const float* rowB[4];
#pragma unroll
  for (int t = 0; t < 4; ++t) {
    int k0 = 4 * t + kb, k1 = k0 + 1;
    int c0 = sClust[wave][k0];  // unconditional LDS reads
    int c1 = sClust[wave][k1];
    int s0 = sSeg[wave][k0];
    int s1 = sSeg[wave][k1];
    int n0 = sNode[wave][k0];
    int n1 = sNode[wave][k1];
    int ok0 = (c0 >= 0), ok1 = (c1 >= 0);
    afrag[t].x = (ok0 && s0 == M) ? 1.0f : 0.0f;
    afrag[t].y = (ok1 && s1 == M) ? 1.0f : 0.0f;
    mskA[t] = ok0 ? 1.0f : 0.0f;
    mskB[t] = ok1 ? 1.0f : 0.0f;
    n0 = ok0 ? n0 : 0;  // clamp pad rows to node 0 (always safe to load)
    n1 = ok1 ? n1 : 0;
    rowA[t] = primalX + (size_t)n0 * kF + n;
    rowB[t] = primalX + (size_t)n1 * kF + n;
  }
  int clRow[8];
#pragma unroll
  for (int j = 0; j < 8; ++j) clRow[j] = sSegCl[wave][rbase + j];

  for (int f = 0; f < 8; ++f) {
    v8f c = {0.f, 0.f, 0.f, 0.f, 0.f, 0.f, 0.f, 0.f};
#pragma unroll
    for (int t = 0; t < 4; ++t) {
      v2f b;
      b.x = mskA[t] * rowA[t][f * 16];  // unconditional global loads, masked by mul
      b.y = mskB[t] * rowB[t][f * 16];
      c = __builtin_amdgcn_wmma_f32_16x16x4_f32(false, afrag[t], false, b,
                                                (short)0, c, false, false);
    }
#pragma unroll
    for (int j = 0; j < 8; ++j) {
      if (clRow[j] >= 0) atomicAdd(&outPX[(size_t)clRow[j] * kF + f * 16 + n], c[j]);
    }
  }
}

__global__ void countsK(const int* __restrict__ cluster, u32* __restrict__ counts) {
  int i = blockIdx.x * blockDim.x + threadIdx.x;
  if (i < kN) atomicAdd(&counts[cluster[i]], 1u);
}
__global__ void primalDivK(float* __restrict__ px, const u32* __restrict__ counts) {
  long long i = (long long)blockIdx.x * blockDim.x + threadIdx.x;
  if (i < (long long)kN * kF) {
    int c = (int)(i >> 7);
    px[i] = px[i] / fmaxf((float)counts[c], 1.0f);
  }
}

// ---------------- new edges + dual pair keys ----------------
__global__ void edgePairK(const int* __restrict__ src, const int* __restrict__ dst,
                          const int* __restrict__ cluster, int* __restrict__ epi0,
                          int* __restrict__ epi1, u64* __restrict__ items) {
  u32 e = blockIdx.x * blockDim.x + threadIdx.x;
  if (e >= kP2E) return;
  if (e < (u32)kE) {
    int cu = cluster[src[e]], cv = cluster[dst[e]];
    bool valid = (cu != cv);
    epi0[e] = valid ? cu : -1;
    epi1[e] = valid ? cv : -1;
    int a = cu < cv ? cu : cv;
    int b = cu < cv ? cv : cu;
    u64 key = valid ? ((u64)(u32)a * (u64)(kN + 1) + (u64)(u32)b)
                    : ((u64)kN * (u64)(kN + 1) + (u64)kN);
    items[e] = (key << 20) | (u64)e;  // lex order preserved; index in low 20 bits
  } else {
    items[e] = PADITEM;
  }
}
__global__ void headK(const u64* __restrict__ items, int* __restrict__ flags) {
  u32 p = blockIdx.x * blockDim.x + threadIdx.x;
  if (p >= kP2E) return;
  flags[p] = (p == 0) ? 1 : (((items[p] >> 20) != (items[p - 1] >> 20)) ? 1 : 0);
}
__global__ void rankK(const u64* __restrict__ items, const int* __restrict__ incl,
                      int* __restrict__ dualCluster) {
  u32 p = blockIdx.x * blockDim.x + threadIdx.x;
  if (p >= kP2E) return;
  u64 it = items[p];
  if (it != PADITEM) {
    u32 idx = (u32)(it & 0xFFFFFull);
    if (idx < (u32)kE) dualCluster[idx] = incl[p] - 1;
  }
}

// ---------------- dual aggregation (async-to-LDS staging) ----------------
__global__ __launch_bounds__(256)
void dualAggAsyncK(const float* __restrict__ dualX, const int* __restrict__ dualCluster,
                   const int* __restrict__ epi0, float* __restrict__ newDualX,
                   u32* __restrict__ dcounts) {
  __shared__ float sTile[16 * kF];  // 16 edges x 128 features = 8KB
  __shared__ int sDC[16];
  __shared__ int sValid[16];
  const int tid = threadIdx.x;
  const int e0 = blockIdx.x * 16;
  if (tid < 16) {
    int e = e0 + tid;
    int dc = dualCluster[e];
    int vld = (epi0[e] >= 0) ? 1 : 0;
    sDC[tid] = dc;
    sValid[tid] = vld;
    if (vld) atomicAdd(&dcounts[dc], 1u);
  }
  // async copy 16 rows of dual_x into LDS: 512 x b128 chunks, 2 per thread
  u32 ldsBase = (u32)(uintptr_t)(&sTile[0]);
#pragma unroll
  for (int q = 0; q < 2; ++q) {
    int c = tid + q * 256;       // chunk 0..511
    int eLocal = c >> 5;         // 32 chunks per row
    int cr = c & 31;
    const float* gp = dualX + (size_t)(e0 + eLocal) * kF + cr * 4;
    u32 ldsOff = ldsBase + (u32)c * 16u;
    asm volatile("global_load_async_to_lds_b128 %0, %1, off" ::"v"(ldsOff),
                 "v"((u64)(uintptr_t)gp)
                 : "memory");
  }
  asm volatile("s_wait_asynccnt 0" ::: "memory");
  __syncthreads();
  const int eLocal = tid >> 4;          // 16 threads per edge
  const int fBase = (tid & 15) * 8;     // 8 features per thread
  if (sValid[eLocal]) {
    int dc = sDC[eLocal];
#pragma unroll
    for (int i = 0; i < 8; ++i) {
      float v = sTile[eLocal * kF + fBase + i];
      atomicAdd(&newDualX[(size_t)dc * kF + fBase + i], v);
    }
  }
}
__global__ void dualDivK(float* __restrict__ dx, const u32* __restrict__ dcounts) {
  long long i = (long long)blockIdx.x * blockDim.x + threadIdx.x;
  if (i < (long long)kE * kF) {
    int c = (int)(i >> 7);
    dx[i] = dx[i] / fmaxf((float)dcounts[c], 1.0f);
  }
}

// ---------------- host-side helpers ----------------
static void bitonicSort(u64* d, u32 n, hipStream_t st) {
  dim3 g(n / 256);
  for (u32 k = 2; k <= n; k <<= 1)
    for (u32 j = k >> 1; j > 0; j >>= 1)
      bitonicStepK<<<g, 256, 0, st>>>(d, j, k);
}
static void scanInclusive(int* in, int* out, int* bsums, int n, hipStream_t st) {
  int nb = (n + 1023) / 1024;
  scanBlocksK<<<nb, 1024, 0, st>>>(in, out, bsums, n);
  scanSumsExclK<<<1, 1024, 0, st>>>(bsums, nb);
  scanAddK<<<nb, 1024, 0, st>>>(out, bsums, n);
}

extern "C" void kernel_launch(void* const* d_in, const int* in_sizes, int n_in,
                              void* d_out, int out_size, void* d_ws, size_t ws_size,
                              hipStream_t stream) {
  const float* primalX = (const float*)d_in[0];
  const float* dualX = (const float*)d_in[1];
  const float* att = (const float*)d_in[2];
  const int* src = (const int*)d_in[3];
  const int* dst = src + kE;

  float* outPX = (float*)d_out;                       // N x F
  float* outDX = outPX + (size_t)kN * kF;             // E x F
  int* epi0 = (int*)(outDX + (size_t)kE * kF);        // E
  int* epi1 = epi0 + kE;                              // E
  int* cluster = epi1 + kE;                           // N

  char* p = (char*)d_ws;
  auto carve = [&](size_t bytes) -> void* {
    void* r = (void*)p;
    p += (bytes + 255) & ~(size_t)255;
    return r;
  };
  u64* items = (u64*)carve((size_t)kP2E * 8);
  int* labels = (int*)carve((size_t)kN * 4);
  int* tmpL = (int*)carve((size_t)kN * 4);
  int* bufA = (int*)carve((size_t)kP2E * 4);
  int* bufB = (int*)carve((size_t)kP2E * 4);
  int* bsums = (int*)carve(1024 * 4);
  unsigned char* pm = (unsigned char*)carve(kE);
  int* dualCluster = (int*)carve((size_t)kE * 4);
  u32* counts = (u32*)carve((size_t)kN * 4);
  u32* dcounts = (u32*)carve((size_t)kE * 4);
  (void)ws_size; (void)n_in; (void)in_sizes; (void)out_size;

  // 0) zero accumulators (must happen every call)
  zeroF<<<cdiv((long long)kN * kF, 256), 256, 0, stream>>>(outPX, (long long)kN * kF);
  zeroF<<<cdiv((long long)kE * kF, 256), 256, 0, stream>>>(outDX, (long long)kE * kF);
  zeroU32<<<cdiv(kN, 256), 256, 0, stream>>>(counts, kN);
  zeroU32<<<cdiv(kE, 256), 256, 0, stream>>>(dcounts, kE);
  zeroU8<<<cdiv(kE, 256), 256, 0, stream>>>(pm, kE);

  // 1) top-k pooled edges by attention (exact stable argsort via bitonic)
  attKeysK<<<kP2E / 256, 256, 0, stream>>>(att, items);
  bitonicSort(items, kP2E, stream);
  poolMaskK<<<cdiv(kPool, 256), 256, 0, stream>>>(items, pm);

  // 2) min-label propagation with pointer jumping
  initLabelsK<<<cdiv(kN, 256), 256, 0, stream>>>(labels);
  for (int it = 0; it < kIters; ++it) {
    relaxK<<<cdiv(kE, 256), 256, 0, stream>>>(labels, src, dst, pm);
    jumpK<<<cdiv(kN, 256), 256, 0, stream>>>(tmpL, labels);
    jumpK<<<cdiv(kN, 256), 256, 0, stream>>>(labels, tmpL);
  }

  // 3) compact labels -> cluster ids (rank of root via prefix scan)
  isRootK<<<cdiv(kN, 256), 256, 0, stream>>>(labels, bufA);
  scanInclusive(bufA, bufB, bsums, kN, stream);
  clusterK<<<cdiv(kN, 256), 256, 0, stream>>>(labels, bufB, cluster);

  // 4) primal feature mean: sort nodes by cluster, WMMA one-hot segment sums
  nodeKeysK<<<kP2N / 256, 256, 0, stream>>>(cluster, items);
  bitonicSort(items, kP2N, stream);
  primalAggWmmaK<<<kP2N / (16 * 8), 256, 0, stream>>>(items, primalX, outPX);
  countsK<<<cdiv(kN, 256), 256, 0, stream>>>(cluster, counts);
  primalDivK<<<cdiv((long long)kN * kF, 256), 256, 0, stream>>>(outPX, counts);

  // 5) rebuild edges + dual pair dedup (sort + head-flag ranks)
  edgePairK<<<kP2E / 256, 256, 0, stream>>>(src, dst, cluster, epi0, epi1, items);
  bitonicSort(items, kP2E, stream);
  headK<<<kP2E / 256, 256, 0, stream>>>(items, bufA);
  scanInclusive(bufA, bufB, bsums, (int)kP2E, stream);
  rankK<<<kP2E / 256, 256, 0, stream>>>(items, bufB, dualCluster);

  // 6) dual feature mean with async-to-LDS staging
  dualAggAsyncK<<<kE / 16, 256, 0, stream>>>(dualX, dualCluster, epi0, outDX, dcounts);
  dualDivK<<<cdiv((long long)kE * kF, 256), 256, 0, stream>>>(outDX, dcounts);
}